// BilinearCNN2D_58231166599669
// MI455X (gfx1250) — compile-verified
//
#include <hip/hip_runtime.h>
#include <hip/hip_bf16.h>
#include <math.h>

// ---------------------------------------------------------------------------
// Bilinear CNN pooling for MI455X (gfx1250):
//   dotted[b] = L[b]^T @ R[b]   (M=N=128, K=112*112=12544, f32)
//   out = l2norm(signed_sqrt(dotted))
// f32 path -> V_WMMA_F32_16X16X4_F32 (exact precision match to reference).
// Global->LDS staging via GLOBAL_LOAD_ASYNC_TO_LDS_B128 (ASYNCcnt) with
// double-buffered LDS so chunk i+1 loads overlap chunk i's 64 WMMAs.
// ---------------------------------------------------------------------------

typedef __attribute__((ext_vector_type(2))) float v2f;
typedef __attribute__((ext_vector_type(8))) float v8f;

#define BATCH   32
#define CDIM    128
#define KDIM    12544        // 112*112
#define KC      32           // K rows staged in LDS per chunk
#define LDSTR   136          // 128 + 8 pad floats (16B aligned rows, no bank conflicts)
#define BUFSZ   (2 * KC * LDSTR)          // floats per (L,R) double entry
#define OUT_PER_BATCH (CDIM*CDIM)         // 16384

// Issue this thread's share of one chunk: 4 x B128 per matrix, async to LDS.
__device__ __forceinline__ void issue_chunk_async(const float* __restrict__ pl,
                                                  const float* __restrict__ pr,
                                                  uint32_t ldsL, uint32_t ldsR,
                                                  int lrow, int lquad) {
#pragma unroll
    for (int j = 0; j < 4; ++j) {
        const uint32_t cf   = (uint32_t)(lquad + 8 * j) * 16u;        // byte col in row
        const uint32_t voff = (uint32_t)lrow * (CDIM * 4u) + cf;      // global byte off
        const uint32_t lofL = ldsL + (uint32_t)lrow * (LDSTR * 4u) + cf;
        const uint32_t lofR = ldsR + (uint32_t)lrow * (LDSTR * 4u) + cf;
        asm volatile("global_load_async_to_lds_b128 %0, %1, %2"
                     :: "v"(lofL), "v"(voff), "s"(pl) : "memory");
        asm volatile("global_load_async_to_lds_b128 %0, %1, %2"
                     :: "v"(lofR), "v"(voff), "s"(pr) : "memory");
    }
}

// ---------------------------------------------------------------------------
// GEMM partial kernel: grid = (32, ksplit), 256 threads (8 waves).
// Workgroup computes full 128x128 for its K-slice; wave w owns a 2x4 block of
// 16x16 tiles (wm = w>>1 selects 32 M-rows, wn = w&1 selects 64 N-cols).
// ---------------------------------------------------------------------------
__global__ __launch_bounds__(256, 2)
void bcnn_gemm_f32(const float* __restrict__ L, const float* __restrict__ R,
                   float* __restrict__ part, int kchunk) {
    const int b = blockIdx.x;        // batch
    const int s = blockIdx.y;        // K slice
    const int t    = threadIdx.x;
    const int lane = t & 31;
    const int w    = t >> 5;
    const int wm   = w >> 1;         // 0..3 -> M block of 32 rows
    const int wn   = w & 1;          // 0..1 -> N block of 64 cols
    const int ll   = lane & 15;
    const int hi   = lane >> 4;

    __shared__ alignas(16) float lds[2 * BUFSZ];     // double buffer: [buf][L|R]
    const uint32_t ldsBase = (uint32_t)(size_t)(void*)&lds[0];  // LDS byte offset

    const size_t base = ((size_t)b * KDIM + (size_t)s * kchunk) * CDIM;
    const float* gl = L + base;
    const float* gr = R + base;

    v8f acc[2][4];
#pragma unroll
    for (int mt = 0; mt < 2; ++mt)
#pragma unroll
        for (int nt = 0; nt < 4; ++nt)
#pragma unroll
            for (int i = 0; i < 8; ++i) acc[mt][nt][i] = 0.0f;

    const int lrow  = t >> 3;        // 0..31 : staged K row this thread loads
    const int lquad = t & 7;         // 8 threads x 4 float4 cover one 128-f row
    const int nchunks = kchunk / KC;

    // Prologue: issue chunk 0 into buffer 0.
    issue_chunk_async(gl, gr,
                      ldsBase, ldsBase + KC * LDSTR * 4u, lrow, lquad);

    for (int it = 0; it < nchunks; ++it) {
        // My async loads for chunk `it` are done; barrier makes everyone's
        // visible AND guarantees all waves finished reading the other buffer
        // before anyone re-issues into it below.
        asm volatile("s_wait_asynccnt 0" ::: "memory");
        __syncthreads();

        const uint32_t bufOff = (uint32_t)(it & 1) * (BUFSZ * 4u);
        if (it + 1 < nchunks) {
            const uint32_t nxtOff = (uint32_t)((it + 1) & 1) * (BUFSZ * 4u);
            const size_t xn = (size_t)(it + 1) * KC * CDIM;
            issue_chunk_async(gl + xn, gr + xn,
                              ldsBase + nxtOff,
                              ldsBase + nxtOff + KC * LDSTR * 4u,
                              lrow, lquad);
        }

        const float* __restrict__ Llds = (const float*)&lds[(it & 1) * BUFSZ];
        const float* __restrict__ Rlds = Llds + KC * LDSTR;
        (void)bufOff;

#pragma unroll
        for (int kk = 0; kk < KC; kk += 4) {
            // ISA 16x4 f32 A layout: lanes 0-15 hold K=kk,kk+1 ; lanes 16-31 K=kk+2,kk+3
            const int kb = kk + 2 * hi;
            v2f a0, a1;                    // two M-tiles of A = L^T (M = channel of L)
            a0[0] = Llds[(kb    ) * LDSTR + wm * 32 + ll];
            a0[1] = Llds[(kb + 1) * LDSTR + wm * 32 + ll];
            a1[0] = Llds[(kb    ) * LDSTR + wm * 32 + 16 + ll];
            a1[1] = Llds[(kb + 1) * LDSTR + wm * 32 + 16 + ll];
#pragma unroll
            for (int nt = 0; nt < 4; ++nt) {
                const int n = wn * 64 + nt * 16 + ll;
                v2f bf;
                bf[0] = Rlds[(kb    ) * LDSTR + n];
                bf[1] = Rlds[(kb + 1) * LDSTR + n];
                acc[0][nt] = __builtin_amdgcn_wmma_f32_16x16x4_f32(
                    false, a0, false, bf, (short)0, acc[0][nt], false, false);
                acc[1][nt] = __builtin_amdgcn_wmma_f32_16x16x4_f32(
                    false, a1, false, bf, (short)0, acc[1][nt], false, false);
            }
        }
    }

    // C/D layout: VGPR i, lanes 0-15 -> M=i, lanes 16-31 -> M=8+i ; N = lane&15
    float* po = part + ((size_t)s * BATCH + b) * OUT_PER_BATCH;
#pragma unroll
    for (int mt = 0; mt < 2; ++mt)
#pragma unroll
        for (int nt = 0; nt < 4; ++nt)
#pragma unroll
            for (int i = 0; i < 8; ++i) {
                const int m = wm * 32 + mt * 16 + i + 8 * hi;
                const int n = wn * 64 + nt * 16 + ll;
                po[m * CDIM + n] = acc[mt][nt][i];
            }
}

// ---------------------------------------------------------------------------
// Finalize: one workgroup per batch. Sum K-slice partials deterministically,
// use  sum(sqrted^2) == sum(|v|) + 16384*EPS_SQRT  to get the norm in one
// reduction pass, then write sign(v)*sqrt(|v|+eps) * rsqrt(max(sum,1e-12)).
// ---------------------------------------------------------------------------
__global__ __launch_bounds__(256)
void bcnn_finalize(const float* __restrict__ part, float* __restrict__ out, int ksplit) {
    const int b = blockIdx.x;
    const int t = threadIdx.x;
    __shared__ float red[256];

    float sumabs = 0.0f;
    for (int i = t; i < OUT_PER_BATCH; i += 256) {
        float v = 0.0f;
        for (int s = 0; s < ksplit; ++s)
            v += part[((size_t)s * BATCH + b) * OUT_PER_BATCH + i];
        out[(size_t)b * OUT_PER_BATCH + i] = v;   // stash raw dotted value
        sumabs += fabsf(v);
    }
    red[t] = sumabs;
    __syncthreads();
    for (int off = 128; off > 0; off >>= 1) {
        if (t < off) red[t] += red[t + off];
        __syncthreads();
    }
    const float sumsq = red[0] + (float)OUT_PER_BATCH * 1e-9f;
    const float inv   = rsqrtf(fmaxf(sumsq, 1e-12f));

    for (int i = t; i < OUT_PER_BATCH; i += 256) {
        const size_t idx = (size_t)b * OUT_PER_BATCH + i;
        const float v  = out[idx];                 // same thread wrote it
        const float sg = (v > 0.0f) ? 1.0f : ((v < 0.0f) ? -1.0f : 0.0f);
        out[idx] = sg * sqrtf(fabsf(v) + 1e-9f) * inv;
    }
}

// ---------------------------------------------------------------------------
extern "C" void kernel_launch(void* const* d_in, const int* in_sizes, int n_in,
                              void* d_out, int out_size, void* d_ws, size_t ws_size,
                              hipStream_t stream) {
    const float* L = (const float*)d_in[0];
    const float* R = (const float*)d_in[1];
    float* out = (float*)d_out;

    const size_t per = (size_t)BATCH * OUT_PER_BATCH * sizeof(float);  // 2 MB

    // Pick largest K split whose partial buffers fit in d_ws; all candidates
    // divide 12544 into multiples of KC=32. Deterministic (depends on ws_size only).
    int ksplit;
    float* part;
    if      (ws_size >= 8 * per) { ksplit = 8; part = (float*)d_ws; }
    else if (ws_size >= 4 * per) { ksplit = 4; part = (float*)d_ws; }
    else if (ws_size >= 2 * per) { ksplit = 2; part = (float*)d_ws; }
    else if (ws_size >= 1 * per) { ksplit = 1; part = (float*)d_ws; }
    else                         { ksplit = 1; part = out; }  // acc in-place in d_out

    const int kchunk = KDIM / ksplit;  // 1568 / 3136 / 6272 / 12544, all % 32 == 0

    dim3 grid(BATCH, ksplit);
    bcnn_gemm_f32<<<grid, 256, 0, stream>>>(L, R, part, kchunk);
    bcnn_finalize<<<BATCH, 256, 0, stream>>>(part, out, ksplit);
}